// ConvGraphEncoder_32341103738939
// MI455X (gfx1250) — compile-verified
//
#include <hip/hip_runtime.h>

typedef float v2f __attribute__((ext_vector_type(2)));
typedef float v8f __attribute__((ext_vector_type(8)));

#define N_NODES 50000
#define N_EDGES 800000
#define D 128            // D_IN == D_HID == 128
#define EPS_F 1e-7f

// ---------------------------------------------------------------------------
// Zero the neighbor accumulator (L2-resident, 25.6 MB)
// ---------------------------------------------------------------------------
__global__ void cge_zero_kernel(float* __restrict__ p, int n) {
    int i = blockIdx.x * blockDim.x + threadIdx.x;
    if (i < n) p[i] = 0.0f;
}

// ---------------------------------------------------------------------------
// Edge scatter: one wave32 per edge. Each lane moves 4 contiguous floats
// (global_load_b128 gather + 4x global_atomic_add_f32). Feature matrix and
// accumulator both fit in the 192 MB L2, so this runs at L2 atomic rate.
// h_stride lets layer-2 read h1 staged inside d_out (stride 256).
// ---------------------------------------------------------------------------
__global__ __launch_bounds__(256) void cge_scatter_kernel(
    const float* __restrict__ h, int h_stride,
    const int* __restrict__ esrc, const int* __restrict__ edst,
    float* __restrict__ neigh) {
    int lane = threadIdx.x & 31;
    int edge = (blockIdx.x * blockDim.x + threadIdx.x) >> 5;
    if (edge >= N_EDGES) return;
    int s = esrc[edge];
    int d = edst[edge];
    float4 v = reinterpret_cast<const float4*>(h + (size_t)s * h_stride)[lane];
    float* np = neigh + (size_t)d * D + lane * 4;
    atomicAdd(np + 0, v.x);
    atomicAdd(np + 1, v.y);
    atomicAdd(np + 2, v.z);
    atomicAdd(np + 3, v.w);
}

// ---------------------------------------------------------------------------
// Fused SAGE layer GEMM using V_WMMA_F32_16X16X4_F32 (full f32 precision):
//   out = relu( concat([A, neigh * invdeg]) @ W + bias )
// One 16x16 C tile per wave; 8 waves per block cover the 8 N-tiles of one
// M-tile (3125 blocks, M = 3125*16 = 50000 exactly; K = 256 = 64 wmma steps).
// Mean-division and concat are fused into the A-operand load.
//
// ISA f32 operand layouts (CDNA5 ISA §7.12.2):
//   A 16x4 : lane&15 = M row; lane>>4 selects K-pair {0,1} vs {2,3} in v[0],v[1]
//   B 4x16 : lane&15 = N col; lane>>4 selects K-pair in v[0],v[1]
//   C 16x16: lane&15 = N; VGPR j holds M = j + 8*(lane>>4)
// ---------------------------------------------------------------------------
__global__ __launch_bounds__(256) void cge_gemm_kernel(
    const float* __restrict__ A, int a_stride,
    const float* __restrict__ neigh,
    const float* __restrict__ deg,
    const float* __restrict__ W,        // [256, 128] row-major
    const float* __restrict__ bias,     // [128]
    float* __restrict__ out, int out_stride) {
    const int lane  = threadIdx.x & 31;
    const int warp  = threadIdx.x >> 5;      // 0..7  -> N tile
    const int mtile = blockIdx.x;            // 0..3124
    const int rc    = lane & 15;             // A-row / B-col within tile
    const int khalf = lane >> 4;             // 0 or 1 -> K-pair select

    const int m = mtile * 16 + rc;           // global row for A operand
    const int n = warp * 16 + rc;            // global col for B operand

    const float invdeg = 1.0f / (deg[m] + EPS_F);
    const float* __restrict__ arow = A     + (size_t)m * a_stride;
    const float* __restrict__ nrow = neigh + (size_t)m * D;

    v8f c = {};

    // K = 0..127 : self features
#pragma unroll 8
    for (int k0 = 0; k0 < 128; k0 += 4) {
        const int k = k0 + 2 * khalf;
        v2f a; a.x = arow[k];              a.y = arow[k + 1];
        v2f b; b.x = W[(size_t)k * D + n]; b.y = W[(size_t)(k + 1) * D + n];
        c = __builtin_amdgcn_wmma_f32_16x16x4_f32(
            false, a, false, b, (short)0, c, false, false);
    }
    // K = 128..255 : mean-aggregated neighbor features (division fused here)
#pragma unroll 8
    for (int k0 = 0; k0 < 128; k0 += 4) {
        const int k = k0 + 2 * khalf;
        v2f a; a.x = nrow[k] * invdeg;     a.y = nrow[k + 1] * invdeg;
        v2f b; b.x = W[(size_t)(128 + k) * D + n];
               b.y = W[(size_t)(129 + k) * D + n];
        c = __builtin_amdgcn_wmma_f32_16x16x4_f32(
            false, a, false, b, (short)0, c, false, false);
    }

    // Epilogue: bias + ReLU, store per C-layout
    const float bn    = bias[n];
    const int   mbase = mtile * 16 + khalf * 8;
#pragma unroll
    for (int j = 0; j < 8; ++j) {
        float v = c[j] + bn;
        v = v > 0.0f ? v : 0.0f;
        out[(size_t)(mbase + j) * out_stride + n] = v;
    }
}

// ---------------------------------------------------------------------------
// Final: out[:, 128:256] = x  (overwrites the h1 staging area)
// ---------------------------------------------------------------------------
__global__ void cge_copyx_kernel(const float* __restrict__ x,
                                 float* __restrict__ out) {
    int i = blockIdx.x * blockDim.x + threadIdx.x;   // over N_NODES*32 float4s
    if (i >= N_NODES * (D / 4)) return;
    int r  = i >> 5;            // node
    int c4 = i & 31;            // float4 column
    float4 v = reinterpret_cast<const float4*>(x)[i];
    reinterpret_cast<float4*>(out + (size_t)r * 256 + 128)[c4] = v;
}

// ---------------------------------------------------------------------------
extern "C" void kernel_launch(void* const* d_in, const int* in_sizes, int n_in,
                              void* d_out, int out_size, void* d_ws, size_t ws_size,
                              hipStream_t stream) {
    const float* x    = (const float*)d_in[0];
    const int*   esrc = (const int*)  d_in[1];
    const int*   edst = (const int*)  d_in[2];
    const float* deg  = (const float*)d_in[3];
    const float* W1   = (const float*)d_in[4];
    const float* b1   = (const float*)d_in[5];
    const float* W2   = (const float*)d_in[6];
    const float* b2   = (const float*)d_in[7];
    float* out = (float*)d_out;

    float* neigh = (float*)d_ws;                 // 50000*128 floats = 25.6 MB
    float* h1    = out + 128;                    // staged in out[:,128:256], stride 256

    const int NF = N_NODES * D;                  // 6,400,000
    dim3 blk(256);
    dim3 zgrid((NF + 255) / 256);                                  // 25000
    dim3 sgrid(((size_t)N_EDGES * 32 + 255) / 256);                // 100000
    dim3 ggrid(N_NODES / 16);                                      // 3125

    // ---- layer 1 ----
    cge_zero_kernel   <<<zgrid, blk, 0, stream>>>(neigh, NF);
    cge_scatter_kernel<<<sgrid, blk, 0, stream>>>(x, D, esrc, edst, neigh);
    cge_gemm_kernel   <<<ggrid, blk, 0, stream>>>(x, D, neigh, deg, W1, b1,
                                                  h1, 256);
    // ---- layer 2 ----
    cge_zero_kernel   <<<zgrid, blk, 0, stream>>>(neigh, NF);
    cge_scatter_kernel<<<sgrid, blk, 0, stream>>>(h1, 256, esrc, edst, neigh);
    cge_gemm_kernel   <<<ggrid, blk, 0, stream>>>(h1, 256, neigh, deg, W2, b2,
                                                  out, 256);
    // ---- out[:, 128:256] = x ----
    cge_copyx_kernel  <<<(N_NODES * (D / 4) + 255) / 256, blk, 0, stream>>>(x, out);
}